// AllAmplitude_23854248362621
// MI455X (gfx1250) — compile-verified
//
#include <hip/hip_runtime.h>

// ===================== compile-time math (mirrors reference) =====================
constexpr double factd(int n){ double r=1.0; for(int i=2;i<=n;++i) r*=double(i); return r; }
constexpr double csqrt_(double x){
  if(x<=0.0) return 0.0;
  double g = x<1.0 ? 1.0 : x;
  for(int i=0;i<200;++i) g = 0.5*(g + x/g);
  return g;
}
constexpr int iabs_(int v){ return v<0?-v:v; }
constexpr double cgc(int j1,int j2,int m1,int m2,int j,int m){
  if(m1+m2!=m || j<iabs_(j1-j2) || j>j1+j2) return 0.0;
  if(iabs_(m1)>j1 || iabs_(m2)>j2 || iabs_(m)>j) return 0.0;
  double pref = csqrt_((2.0*j+1.0)*factd(j+j1-j2)*factd(j-j1+j2)*factd(j1+j2-j)/factd(j1+j2+j+1));
  pref *= csqrt_(factd(j+m)*factd(j-m)*factd(j1-m1)*factd(j1+m1)*factd(j2-m2)*factd(j2+m2));
  double tot=0.0;
  for(int k=0;k<=j1+j2-j;++k){
    int d0=k,d1=j1+j2-j-k,d2=j1-m1-k,d3=j2+m2-k,d4=j-j2+m1+k,d5=j-j1-m2+k;
    if(d0<0||d1<0||d2<0||d3<0||d4<0||d5<0) continue;
    double p=factd(d0)*factd(d1)*factd(d2)*factd(d3)*factd(d4)*factd(d5);
    tot += ((k&1)? -1.0:1.0)/p;
  }
  return pref*tot;
}
// fac = sqrt((2l+1)/(2J1+1)) * cg(J2,J3,l2,-l3,s,l2-l3) * cg(l,s,0,l2-l3,J1,l2-l3)
constexpr double dfac(int J1,int J2,int J3,int l,int s,int l2,int l3){
  return csqrt_((2.0*l+1.0)/(2.0*J1+1.0))
       * cgc(J2,J3,l2,-l3,s,l2-l3)
       * cgc(l,s,0,l2-l3,J1,l2-l3);
}

constexpr double d_ECOM=4.59925, d_MB=2.01026, d_MC=0.13957061, d_MD=2.00685;
constexpr double bar2d(double z){ return 9.0+(3.0+z)*z; }
constexpr double bar4d(double z){ return z*(z*(z*(z+10.0)+135.0)+1575.0)+11025.0; }
constexpr double getqsd(double m,double m1,double m2){
  double ms=m1+m2, md=m1-m2;
  return csqrt_((m+ms)*(m-ms)*(m+md)*(m-md))/(2.0*m);
}

// Zc_4025: ch=-1, Jr=1, m0=4.026, g0=0.025
constexpr double ZC_M0=4.026, ZC_G0=0.025;
constexpr double ZC_Q0RT  = getqsd(d_ECOM, d_MC, ZC_M0);
constexpr double ZC_B2Z0RT= bar2d(9.0*ZC_Q0RT*ZC_Q0RT);
constexpr double ZC_Q0ND  = getqsd(ZC_M0, d_MB, d_MC);   // reference quirk: (M0_B, M0_C)
constexpr double ZC_B2Z0ND= bar2d(9.0*ZC_Q0ND*ZC_Q0ND);
constexpr double ZC_Q0BW  = getqsd(ZC_M0, d_MB, d_MD);   // BW q0, L=0
// D2_2460: ch=1, Jr=2, m0=2.4607, g0=0.0475
constexpr double D2_M0=2.4607, D2_G0=0.0475;
constexpr double D2_Q0RT  = getqsd(d_ECOM, d_MD, D2_M0);
constexpr double D2_B2Z0RT= bar2d(9.0*D2_Q0RT*D2_Q0RT);
constexpr double D2_B4Z0RT= bar4d(9.0*D2_Q0RT*D2_Q0RT);
constexpr double D2_Q0ND  = getqsd(D2_M0, d_MB, d_MC);
constexpr double D2_B2Z0ND= bar2d(9.0*D2_Q0ND*D2_Q0ND);  // BW L=2 uses same q0
// D1_2430: ch=121, Jr=1, m0=2.427, g0=0.384
constexpr double D1_M0=2.427, D1_G0=0.384;
constexpr double D1_Q0RT  = getqsd(d_ECOM, d_MB, D1_M0);
constexpr double D1_B2Z0RT= bar2d(9.0*D1_Q0RT*D1_Q0RT);
constexpr double D1_Q0ND  = getqsd(D1_M0, d_MC, d_MD);
constexpr double D1_B2Z0ND= bar2d(9.0*D1_Q0ND*D1_Q0ND);  // BW L=0

// coefficient tables
struct T2x3 { double c[2][3]; };
constexpr T2x3 mkZCF1(){ // dynamic_f(J1=1,J2=1,J3=0, l2=r, l3=0), ls={(0,1),(2,1)}
  T2x3 t{}; const int ls[2][2]={{0,1},{2,1}};
  for(int i=0;i<2;++i) for(int r=-1;r<=1;++r) t.c[i][r+1]=dfac(1,1,0,ls[i][0],ls[i][1],r,0);
  return t;
}
constexpr T2x3 ZCF1 = mkZCF1();

struct T3x3x3 { double c[3][3][3]; };
constexpr T3x3x3 mkT111(){ // (J1,J2,J3)=(1,1,1), ls={(0,1),(2,1),(2,2)} (Zc F2 and D1 F1)
  T3x3x3 t{}; const int ls[3][2]={{0,1},{2,1},{2,2}};
  for(int i=0;i<3;++i) for(int p=-1;p<=1;++p) for(int w=-1;w<=1;++w)
    t.c[i][p+1][w+1]=dfac(1,1,1,ls[i][0],ls[i][1],p,w);
  return t;
}
constexpr T3x3x3 T111 = mkT111();

struct T5x5x3 { double c[5][5][3]; };
constexpr T5x5x3 mkD2F1(){ // (J1,J2,J3)=(1,2,1), ls={(0,1),(2,1),(2,2),(2,3),(4,3)}
  T5x5x3 t{}; const int ls[5][2]={{0,1},{2,1},{2,2},{2,3},{4,3}};
  for(int i=0;i<5;++i) for(int r=-2;r<=2;++r) for(int f=-1;f<=1;++f)
    t.c[i][r+2][f+1]=dfac(1,2,1,ls[i][0],ls[i][1],r,f);
  return t;
}
constexpr T5x5x3 D2F1 = mkD2F1();

struct T3 { double c[3]; };
constexpr T3 mkD2F2(){ // (J1,J2,J3)=(2,1,0), ls={(2,1)}
  T3 t{}; for(int p=-1;p<=1;++p) t.c[p+1]=dfac(2,1,0,2,1,p,0);
  return t;
}
constexpr T3 D2F2 = mkD2F2();

constexpr double D1F2C0 = dfac(1,0,1,0,1,0,0); // only w=0 survives (J2=0 kills w=+-1)
constexpr double D1F2C2 = dfac(1,0,1,2,1,0,0);

// ===================== device helpers =====================
struct cf { float x, y; };
__device__ __forceinline__ cf cadd(cf a, cf b){ return {a.x+b.x, a.y+b.y}; }
__device__ __forceinline__ cf cmul(cf a, cf b){ return {a.x*b.x - a.y*b.y, a.x*b.y + a.y*b.x}; }
__device__ __forceinline__ cf csca(cf a, float s){ return {a.x*s, a.y*s}; }
__device__ __forceinline__ cf cconj(cf a){ return {a.x, -a.y}; }
__device__ __forceinline__ void cacc(cf& d, cf a, cf b){
  d.x = fmaf(a.x, b.x, fmaf(-a.y, b.y, d.x));
  d.y = fmaf(a.x, b.y, fmaf( a.y, b.x, d.y));
}
__device__ __forceinline__ cf epmi(float ang){ float s,c; sincosf(ang,&s,&c); return {c,-s}; } // e^{-i ang}
__device__ __forceinline__ cf crecip(float re, float im){
  float inv = 1.f/fmaf(re,re,im*im); return {re*inv, -im*inv};
}
__device__ __forceinline__ float getqf(float m, float m1, float m2){
  float ms=m1+m2, md=m1-m2;
  return sqrtf((m+ms)*(m-ms)*(m+md)*(m-md))/(2.f*m);
}
__device__ __forceinline__ float B2f(float z){ return 9.f + (3.f+z)*z; }
__device__ __forceinline__ float B4f(float z){ return fmaf(z, fmaf(z, fmaf(z, z+10.f, 135.f), 1575.f), 11025.f); }

// Wigner small-d, j=1: d[m1+1][m2+1]
__device__ __forceinline__ void d1mat(float c, float d[3][3]){
  float h1 = fminf(fmaxf(0.5f*(1.f+c),0.f),1.f);
  float h2 = fminf(fmaxf(0.5f*(1.f-c),0.f),1.f);
  float c2=sqrtf(h1), s2=sqrtf(h2);
  float cs = 1.4142135623730951f*c2*s2;     // sin(b)/sqrt(2)
  d[0][0]=h1;  d[0][1]=cs;  d[0][2]=h2;
  d[1][0]=-cs; d[1][1]=c;   d[1][2]=cs;
  d[2][0]=h2;  d[2][1]=-cs; d[2][2]=h1;
}
// Wigner small-d, j=2, m1 in [-2,2] rows, m2 in [-1,1] cols
__device__ __forceinline__ void d2mat(float c, float d[5][3]){
  float h1 = fminf(fmaxf(0.5f*(1.f+c),0.f),1.f);
  float h2 = fminf(fmaxf(0.5f*(1.f-c),0.f),1.f);
  float c2=sqrtf(h1), s2=sqrtf(h2);
  float s = 2.f*c2*s2;
  const float R64 = 0.6123724356957945f;    // sqrt(6)/4
  const float K15 = 1.224744871391589f;     // sqrt(3/2)
  float ssq = s*s, ksc = K15*s*c;
  d[0][0]=h1*s;             d[0][1]=R64*ssq;            d[0][2]=h2*s;             // m1=-2
  d[1][0]=h1*(2.f*c-1.f);   d[1][1]=ksc;                d[1][2]=h2*(2.f*c+1.f);   // m1=-1
  d[2][0]=-ksc;             d[2][1]=0.5f*(3.f*c*c-1.f); d[2][2]=ksc;              // m1=0
  d[3][0]=h2*(2.f*c+1.f);   d[3][1]=-ksc;               d[3][2]=h1*(2.f*c-1.f);   // m1=1
  d[4][0]=-h2*s;            d[4][1]=R64*ssq;            d[4][2]=-h1*s;            // m1=2
}

// wave32 sum reduction: one V_WMMA_F32_16X16X4_F32 (A=ones, B=partials -> column sums),
// then 4 xor-shuffles. Deterministic, exact fp32.
typedef float v2f_t __attribute__((ext_vector_type(2)));
typedef float v8f_t __attribute__((ext_vector_type(8)));
__device__ __forceinline__ float wave_reduce(float v){
#if defined(__gfx1250__) && __has_builtin(__builtin_amdgcn_wmma_f32_16x16x4_f32)
  v2f_t a; a[0]=1.f; a[1]=1.f;    // A = ones(16x4)
  v2f_t b; b[0]=v;   b[1]=0.f;    // B holds each lane's partial exactly once
  v8f_t c = {0.f,0.f,0.f,0.f,0.f,0.f,0.f,0.f};
  c = __builtin_amdgcn_wmma_f32_16x16x4_f32(false, a, false, b, (short)0, c, false, false);
  float t = c[0];                 // column sum of B (distinct column per lane within each half)
  t += __shfl_xor(t, 1, 32);
  t += __shfl_xor(t, 2, 32);
  t += __shfl_xor(t, 4, 32);
  t += __shfl_xor(t, 8, 32);
  return t;
#else
  for(int m=16;m>=1;m>>=1) v += __shfl_xor(v, m, 32);
  return v;
#endif
}

struct KArgs { const float* p[28]; float* partial; int N; };

// ===================== stage 1: per-event |amp|^2, per-block partial sums =====================
__global__ __launch_bounds__(256) void amp_stage1(KArgs A){
  const int i = blockIdx.x*blockDim.x + threadIdx.x;
  float val = 0.f;
  if(i < A.N){
    const float* gp = A.p[27];
    cf G[16];
    #pragma unroll
    for(int k=0;k<16;++k){ G[k].x = gp[2*k]; G[k].y = gp[2*k+1]; }

    cf amp[2][3][3];
    #pragma unroll
    for(int a=0;a<2;++a)
      #pragma unroll
      for(int e=0;e<3;++e)
        #pragma unroll
        for(int f=0;f<3;++f) amp[a][e][f] = cf{0.f,0.f};

    // ===================== Zc_4025 (BD chain) =====================
    {
      const float m    = A.p[8][i];
      const float phiA = A.p[9][i];
      const float cA   = A.p[10][i];
      const float phiR = A.p[11][i];
      const float cR   = A.p[12][i];
      const float alB  = A.p[13][i];
      const float cbB  = A.p[14][i];
      const float gaB  = A.p[15][i];
      const float alD  = A.p[16][i];
      const float cbD  = A.p[17][i];
      const float gaD  = A.p[18][i];

      const float q_rt = getqf((float)d_ECOM, (float)d_MC, m);
      const float zrt  = 9.f*q_rt*q_rt;
      const float Q2rt = q_rt*q_rt*sqrtf((float)ZC_B2Z0RT / B2f(zrt));

      cf F1[3];
      #pragma unroll
      for(int r=0;r<3;++r)
        F1[r] = cadd(csca(G[0], (float)ZCF1.c[0][r]),
                     csca(G[1], (float)ZCF1.c[1][r]*Q2rt));

      const float q_nd = getqf(m, (float)d_MB, (float)d_MD);
      const float znd  = 9.f*q_nd*q_nd;
      const float Q2nd = q_nd*q_nd*sqrtf((float)ZC_B2Z0ND / B2f(znd));

      cf F2[3][3];
      #pragma unroll
      for(int p=0;p<3;++p)
        #pragma unroll
        for(int w=0;w<3;++w)
          F2[p][w] = cadd(csca(G[2], (float)T111.c[0][p][w]),
                     cadd(csca(G[3], (float)T111.c[1][p][w]*Q2nd),
                          csca(G[4], (float)T111.c[2][p][w]*Q2nd)));

      float dA[3][3]; d1mat(cA, dA);
      float dR[3][3]; d1mat(cR, dR);
      float dB[3][3]; d1mat(cbB, dB);
      float dD[3][3]; d1mat(cbD, dD);

      const cf EA = epmi(phiA);
      const cf ER = epmi(phiR);
      const cf ERm[3] = { cconj(ER), {1.f,0.f}, ER };
      const cf PaB = epmi(alB), PgB = epmi(gaB);
      const cf Pp[3] = { cconj(PaB), {1.f,0.f}, PaB };
      const cf Pe[3] = { cconj(PgB), {1.f,0.f}, PgB };
      const cf PaD = epmi(alD), PgD = epmi(gaD);
      const cf Qw[3] = { cconj(PaD), {1.f,0.f}, PaD };
      const cf Qf[3] = { cconj(PgD), {1.f,0.f}, PgD };

      cf DBf[3][3], DDf[3][3];
      #pragma unroll
      for(int p=0;p<3;++p)
        #pragma unroll
        for(int e=0;e<3;++e)
          DBf[p][e] = csca(cmul(Pp[p],Pe[e]), dB[p][e]);
      #pragma unroll
      for(int w=0;w<3;++w)
        #pragma unroll
        for(int f=0;f<3;++f)
          DDf[w][f] = csca(cmul(Qw[w],Qf[f]), dD[w][f]);

      // BW, L=0, q = getq(m_BD, M0_B, M0_D) == q_nd
      const float width = (float)ZC_G0 * (q_nd*(float)(1.0/ZC_Q0BW)) * ((float)ZC_M0/m);
      const cf bw = crecip(fmaf(m,m,-(float)(ZC_M0*ZC_M0)), (float)ZC_M0*width);
      const cf PhA[2] = { cconj(EA), EA };   // a=-1, a=+1

      #pragma unroll
      for(int r=0;r<3;++r){
        const cf Kr = cmul(bw, cmul(F1[r], ERm[r]));
        cf Gef[3][3];
        #pragma unroll
        for(int e=0;e<3;++e)
          #pragma unroll
          for(int f=0;f<3;++f) Gef[e][f] = cf{0.f,0.f};
        #pragma unroll
        for(int p=0;p<3;++p){
          cf Zf[3] = {{0.f,0.f},{0.f,0.f},{0.f,0.f}};
          #pragma unroll
          for(int w=0;w<3;++w){
            const int dmi = p - w + 1;           // (p - w) + 1, need 0..2
            if(dmi < 0 || dmi > 2) continue;
            const cf cpw = csca(F2[p][w], dR[r][dmi]);
            #pragma unroll
            for(int f=0;f<3;++f) cacc(Zf[f], cpw, DDf[w][f]);
          }
          #pragma unroll
          for(int e=0;e<3;++e)
            #pragma unroll
            for(int f=0;f<3;++f) cacc(Gef[e][f], DBf[p][e], Zf[f]);
        }
        #pragma unroll
        for(int a=0;a<2;++a){
          const cf Wa = csca(cmul(Kr, PhA[a]), dA[2*a][r]);
          #pragma unroll
          for(int e=0;e<3;++e)
            #pragma unroll
            for(int f=0;f<3;++f) cacc(amp[a][e][f], Wa, Gef[e][f]);
        }
      }
    }

    // ===================== D2_2460 (BC chain) =====================
    {
      const float m    = A.p[0][i];
      const float phiA = A.p[1][i];
      const float cA   = A.p[2][i];
      const float phiR = A.p[3][i];
      const float cR   = A.p[4][i];
      const float alB  = A.p[5][i];
      const float cbB  = A.p[6][i];
      const float gaB  = A.p[7][i];

      const float q_rt = getqf((float)d_ECOM, (float)d_MD, m);
      const float zrt  = 9.f*q_rt*q_rt;
      const float q2   = q_rt*q_rt;
      const float Q2rt = q2*sqrtf((float)D2_B2Z0RT / B2f(zrt));
      const float Q4rt = q2*q2*sqrtf((float)D2_B4Z0RT / B4f(zrt));

      cf F1[5][3];
      #pragma unroll
      for(int r=0;r<5;++r)
        #pragma unroll
        for(int f=0;f<3;++f){
          cf t = csca(G[5], (float)D2F1.c[0][r][f]);
          t = cadd(t, csca(G[6], (float)D2F1.c[1][r][f]*Q2rt));
          t = cadd(t, csca(G[7], (float)D2F1.c[2][r][f]*Q2rt));
          t = cadd(t, csca(G[8], (float)D2F1.c[3][r][f]*Q2rt));
          t = cadd(t, csca(G[9], (float)D2F1.c[4][r][f]*Q4rt));
          F1[r][f] = t;
        }

      const float q_nd = getqf(m, (float)d_MB, (float)d_MC);
      const float znd  = 9.f*q_nd*q_nd;
      const float B2z  = B2f(znd);
      const float Q2nd = q_nd*q_nd*sqrtf((float)D2_B2Z0ND / B2z);

      cf F2p[3];
      #pragma unroll
      for(int p=0;p<3;++p) F2p[p] = csca(G[10], (float)D2F2.c[p]*Q2nd);

      float dA[3][3]; d1mat(cA, dA);
      float d2R[5][3]; d2mat(cR, d2R);
      float dB[3][3]; d1mat(cbB, dB);

      const cf EA = epmi(phiA);
      const cf ER = epmi(phiR);
      const cf ER2 = cmul(ER,ER);
      const cf ERm[5] = { cconj(ER2), cconj(ER), {1.f,0.f}, ER, ER2 };
      const cf PaB = epmi(alB), PgB = epmi(gaB);
      const cf Pp[3] = { cconj(PaB), {1.f,0.f}, PaB };
      const cf Pe[3] = { cconj(PgB), {1.f,0.f}, PgB };

      // BW, L=2
      const float rq  = q_nd*(float)(1.0/D2_Q0ND);
      const float rq2 = rq*rq;
      const float rq5 = rq2*rq2*rq;
      const float width = (float)D2_G0 * rq5 * ((float)D2_M0/m) * ((float)D2_B2Z0ND / B2z);
      const cf bw = crecip(fmaf(m,m,-(float)(D2_M0*D2_M0)), (float)D2_M0*width);
      const cf BA[2] = { cmul(bw, cconj(EA)), cmul(bw, EA) };

      cf T[3][3];
      #pragma unroll
      for(int p=0;p<3;++p)
        #pragma unroll
        for(int e=0;e<3;++e)
          T[p][e] = cmul(F2p[p], csca(cmul(Pp[p],Pe[e]), dB[p][e]));

      cf V[5][3];
      #pragma unroll
      for(int r=0;r<5;++r)
        #pragma unroll
        for(int e=0;e<3;++e){
          cf u = {0.f,0.f};
          #pragma unroll
          for(int p=0;p<3;++p){
            u.x = fmaf(T[p][e].x, d2R[r][p], u.x);
            u.y = fmaf(T[p][e].y, d2R[r][p], u.y);
          }
          V[r][e] = cmul(u, ERm[r]);
        }

      #pragma unroll
      for(int a=0;a<2;++a)
        #pragma unroll
        for(int f=0;f<3;++f)
          #pragma unroll
          for(int r=0;r<5;++r){
            const int dmi = r - f;               // (r-2)-(f-1)+1, need 0..2
            if(dmi < 0 || dmi > 2) continue;
            const cf S = csca(cmul(BA[a], F1[r][f]), dA[2*a][dmi]);
            #pragma unroll
            for(int e=0;e<3;++e) cacc(amp[a][e][f], S, V[r][e]);
          }
    }

    // ===================== D1_2430 (CD chain) =====================
    {
      const float m    = A.p[19][i];
      const float phiA = A.p[20][i];
      const float cA   = A.p[21][i];
      const float phiR = A.p[22][i];
      const float cR   = A.p[23][i];
      const float cbD  = A.p[25][i];
      const float gaD  = A.p[26][i];
      // alpha_D_CD unused: only w=0 survives (F2 vanishes for w != 0)

      const float q_rt = getqf((float)d_ECOM, (float)d_MB, m);
      const float Q2rt = q_rt*q_rt*sqrtf((float)D1_B2Z0RT / B2f(9.f*q_rt*q_rt));
      const float q_nd = getqf(m, (float)d_MC, (float)d_MD);
      const float Q2nd = q_nd*q_nd*sqrtf((float)D1_B2Z0ND / B2f(9.f*q_nd*q_nd));

      cf F1[3][3];
      #pragma unroll
      for(int r=0;r<3;++r)
        #pragma unroll
        for(int e=0;e<3;++e)
          F1[r][e] = cadd(csca(G[11], (float)T111.c[0][r][e]),
                     cadd(csca(G[12], (float)T111.c[1][r][e]*Q2rt),
                          csca(G[13], (float)T111.c[2][r][e]*Q2rt)));

      const cf F20 = cadd(csca(G[14], (float)D1F2C0),
                          csca(G[15], (float)D1F2C2*Q2nd));

      float dA[3][3]; d1mat(cA, dA);
      float dR[3][3]; d1mat(cR, dR);
      float dD[3][3]; d1mat(cbD, dD);

      const cf EA = epmi(phiA);
      const cf ER = epmi(phiR);
      const cf ERm[3] = { cconj(ER), {1.f,0.f}, ER };
      const cf PgD = epmi(gaD);
      const cf Qf[3] = { cconj(PgD), {1.f,0.f}, PgD };

      // BW, L=0
      const float width = (float)D1_G0 * (q_nd*(float)(1.0/D1_Q0ND)) * ((float)D1_M0/m);
      const cf bw = crecip(fmaf(m,m,-(float)(D1_M0*D1_M0)), (float)D1_M0*width);
      const cf BA[2] = { cmul(bw, cconj(EA)), cmul(bw, EA) };

      cf Vr[3];
      #pragma unroll
      for(int r=0;r<3;++r) Vr[r] = csca(ERm[r], dR[r][1]);     // w=0 column
      cf CP[3];
      #pragma unroll
      for(int f=0;f<3;++f) CP[f] = csca(cmul(F20, Qf[f]), dD[1][f]); // w=0 row

      cf FV[3][3];
      #pragma unroll
      for(int r=0;r<3;++r)
        #pragma unroll
        for(int e=0;e<3;++e) FV[r][e] = cmul(F1[r][e], Vr[r]);

      #pragma unroll
      for(int a=0;a<2;++a)
        #pragma unroll
        for(int e=0;e<3;++e){
          cf wsum = {0.f,0.f};
          #pragma unroll
          for(int r=0;r<3;++r){
            const int dmi = r - e + 1;          // (r-e)+1, need 0..2
            if(dmi < 0 || dmi > 2) continue;
            const float dv = dA[2*a][dmi];
            wsum.x = fmaf(FV[r][e].x, dv, wsum.x);
            wsum.y = fmaf(FV[r][e].y, dv, wsum.y);
          }
          const cf BW = cmul(BA[a], wsum);
          #pragma unroll
          for(int f=0;f<3;++f) cacc(amp[a][e][f], BW, CP[f]);
        }
    }

    float s = 0.f;
    #pragma unroll
    for(int a=0;a<2;++a)
      #pragma unroll
      for(int e=0;e<3;++e)
        #pragma unroll
        for(int f=0;f<3;++f)
          s = fmaf(amp[a][e][f].x, amp[a][e][f].x,
              fmaf(amp[a][e][f].y, amp[a][e][f].y, s));
    val = s;
  }

  // deterministic block reduction (WMMA wave reduce + fixed-order LDS combine)
  const float wsum = wave_reduce(val);
  __shared__ float sh[8];
  const int lane = threadIdx.x & 31, wid = threadIdx.x >> 5;
  if(lane==0) sh[wid] = wsum;
  __syncthreads();
  if(threadIdx.x==0){
    float t = sh[0];
    #pragma unroll
    for(int w=1;w<8;++w) t += sh[w];
    A.partial[blockIdx.x] = t;
  }
}

// ===================== stage 2: deterministic final reduction =====================
__global__ __launch_bounds__(256) void amp_stage2(const float* __restrict__ partial, int n,
                                                  float* __restrict__ out){
  float v = 0.f;
  for(int i=threadIdx.x; i<n; i+=256) v += partial[i];
  const float wsum = wave_reduce(v);
  __shared__ float sh[8];
  const int lane = threadIdx.x & 31, wid = threadIdx.x >> 5;
  if(lane==0) sh[wid] = wsum;
  __syncthreads();
  if(threadIdx.x==0){
    float t = sh[0];
    #pragma unroll
    for(int w=1;w<8;++w) t += sh[w];
    out[0] = t;
  }
}

extern "C" void kernel_launch(void* const* d_in, const int* in_sizes, int n_in,
                              void* d_out, int out_size, void* d_ws, size_t ws_size,
                              hipStream_t stream) {
  KArgs A;
  for(int k=0;k<28;++k) A.p[k] = (const float*)d_in[k];
  A.N = in_sizes[0];
  A.partial = (float*)d_ws;
  const int NB = (A.N + 255)/256;
  amp_stage1<<<NB, 256, 0, stream>>>(A);
  amp_stage2<<<1, 256, 0, stream>>>((const float*)d_ws, NB, (float*)d_out);
}